// STULayer3D_48550310314360
// MI455X (gfx1250) — compile-verified
//
#include <hip/hip_runtime.h>

// ---------------- problem constants ----------------
#define B_   2
#define V_   2
#define D_   32
#define L_   8
#define X_   32
#define Y_   32
#define Z_   32
#define ZR_  17
#define XYZ_ (X_*Y_*Z_)       // 32768
#define NF_  (X_*Y_*ZR_)      // 17408
#define BV_  (B_*V_)          // 4
#define LH_  (L_*D_)          // 256
#define NTOT_  ((size_t)BV_*D_*XYZ_)   // 4,194,304 floats
#define NFREQ_ ((size_t)BV_*D_*NF_)    // 2,228,224 float2

typedef float v2f __attribute__((ext_vector_type(2)));
typedef float v8f __attribute__((ext_vector_type(8)));
typedef unsigned int u32x4 __attribute__((ext_vector_type(4)));
typedef unsigned int u32x8 __attribute__((ext_vector_type(8)));

__device__ __forceinline__ v8f wmma4(v2f a, v2f b, v8f c) {
  // D = A(16x4,f32) * B(4x16,f32) + C(16x16,f32)
  return __builtin_amdgcn_wmma_f32_16x16x4_f32(false, a, false, b, (short)0, c,
                                               false, false);
}

// ---------------- Tensor Data Mover: 2D tile -> LDS ----------------
// D# per CDNA5 ISA. 2D mode (groups 2/3 NULL).
// All inputs must be wave-uniform; descriptor is forced into SGPRs.
__device__ __forceinline__ void tdm_load_2d(unsigned int lds_byte, unsigned long long gaddr,
                                            unsigned int data_size_code,   // 0=1B,1=2B,2=4B,3=8B
                                            unsigned int tile_d0, unsigned int tile_d1,
                                            unsigned long long tensor_d0, unsigned int tensor_d1,
                                            unsigned long long stride0) {  // data_size units
  u32x4 g0;
  g0.x = __builtin_amdgcn_readfirstlane(1u);                 // count=1, user mode
  g0.y = __builtin_amdgcn_readfirstlane(lds_byte);           // lds_addr
  g0.z = __builtin_amdgcn_readfirstlane((unsigned int)gaddr);// global_addr[31:0]
  g0.w = __builtin_amdgcn_readfirstlane(
      (unsigned int)((gaddr >> 32) & 0x1FFFFFFull) | (2u << 30)); // addr[56:32] | type=2
  u32x8 g1;
  g1[0] = __builtin_amdgcn_readfirstlane(data_size_code << 16);   // wg_mask=0, data_size
  g1[1] = __builtin_amdgcn_readfirstlane((unsigned int)((tensor_d0 & 0xFFFFull) << 16));
  g1[2] = __builtin_amdgcn_readfirstlane(
      (unsigned int)((tensor_d0 >> 16) & 0xFFFFull) | (tensor_d1 << 16));
  g1[3] = __builtin_amdgcn_readfirstlane(((tensor_d1 >> 16) & 0xFFFFu) | (tile_d0 << 16));
  g1[4] = __builtin_amdgcn_readfirstlane(tile_d1 & 0xFFFFu);      // tile_dim2 = 0
  g1[5] = __builtin_amdgcn_readfirstlane((unsigned int)(stride0 & 0xFFFFFFFFull));
  g1[6] = __builtin_amdgcn_readfirstlane((unsigned int)((stride0 >> 32) & 0xFFFFull));
  g1[7] = __builtin_amdgcn_readfirstlane(0u);
  asm volatile("tensor_load_to_lds %0, %1" :: "s"(g0), "s"(g1) : "memory");
}

// ---------------- 32-point FFT, fully register-resident ----------------
__device__ __constant__ const int BR5[32] = {
  0,16,8,24,4,20,12,28,2,18,10,26,6,22,14,30,
  1,17,9,25,5,21,13,29,3,19,11,27,7,23,15,31};

template<int LEN>
__device__ __forceinline__ void fft_stage(float2 v[32], const float* tc, const float* ts) {
  #pragma unroll
  for (int i = 0; i < 32; i += LEN) {
    #pragma unroll
    for (int j = 0; j < LEN/2; ++j) {
      const float wr = tc[j*(32/LEN)];
      const float wi = ts[j*(32/LEN)];
      float2 u = v[i+j];
      float2 q = v[i+j+LEN/2];
      float2 t = make_float2(wr*q.x - wi*q.y, wr*q.y + wi*q.x);
      v[i+j]        = make_float2(u.x + t.x, u.y + t.y);
      v[i+j+LEN/2]  = make_float2(u.x - t.x, u.y - t.y);
    }
  }
}

// SIGN = -1 : forward (e^{-i...}),  SIGN = +1 : inverse (no 1/N scaling here)
template<int SIGN>
__device__ __forceinline__ void fft32(float2 v[32]) {
  float tc[16], ts[16];
  #pragma unroll
  for (int k = 0; k < 16; ++k) {
    float a = -6.28318530717958647692f * (float)k / 32.0f;
    tc[k] = __cosf(a);
    ts[k] = __sinf(a) * (SIGN < 0 ? 1.0f : -1.0f);
  }
  #pragma unroll
  for (int i = 0; i < 32; ++i) {
    const int j = BR5[i];
    if (j > i) { float2 t = v[i]; v[i] = v[j]; v[j] = t; }
  }
  fft_stage<2>(v, tc, ts);
  fft_stage<4>(v, tc, ts);
  fft_stage<8>(v, tc, ts);
  fft_stage<16>(v, tc, ts);
  fft_stage<32>(v, tc, ts);
}

// ---------------- K1: LayerNorm over D ----------------
__global__ __launch_bounds__(256) void ln_kernel(const float* __restrict__ x,
                                                 const float* __restrict__ w,
                                                 const float* __restrict__ b,
                                                 float* __restrict__ z) {
  int idx = blockIdx.x * blockDim.x + threadIdx.x;   // BV_*XYZ_ = 131072
  int bv = idx >> 15;
  int sp = idx & (XYZ_ - 1);
  const float* px = x + (size_t)bv * D_ * XYZ_ + sp;
  float*       pz = z + (size_t)bv * D_ * XYZ_ + sp;
  float vals[D_]; float mu = 0.f;
  #pragma unroll
  for (int d = 0; d < D_; ++d) { vals[d] = px[(size_t)d * XYZ_]; mu += vals[d]; }
  mu *= (1.0f / D_);
  float var = 0.f;
  #pragma unroll
  for (int d = 0; d < D_; ++d) { float t = vals[d] - mu; var += t * t; }
  var *= (1.0f / D_);
  float r = rsqrtf(var + 1e-5f);
  #pragma unroll
  for (int d = 0; d < D_; ++d)
    pz[(size_t)d * XYZ_] = (vals[d] - mu) * r * w[d] + b[d];
}

// ---------------- K2: forward rFFT along Z ----------------
__global__ __launch_bounds__(256) void fftz_fwd(const float* __restrict__ z,
                                                float2* __restrict__ Xf) {
  int line = blockIdx.x * blockDim.x + threadIdx.x;  // BV_*D_*X_*Y_ = 131072
  const float* p = z + (size_t)line * Z_;
  float2 v[32];
  #pragma unroll
  for (int i = 0; i < 32; ++i) v[i] = make_float2(p[i], 0.f);
  fft32<-1>(v);
  float2* q = Xf + (size_t)line * ZR_;
  #pragma unroll
  for (int k = 0; k < ZR_; ++k) q[k] = v[k];
}

// ---------------- K3/K7: complex FFT along Y (in place) ----------------
template<int SIGN>
__global__ __launch_bounds__(256) void ffty_pass(float2* __restrict__ data) {
  int t  = blockIdx.x * blockDim.x + threadIdx.x;    // BV_*D_*X_*ZR_ = 69632
  int zr = t % ZR_;
  int xx = t / ZR_;                                  // (bvd*X_ + x)
  float2* p = data + (size_t)xx * (Y_ * ZR_) + zr;
  float2 v[32];
  #pragma unroll
  for (int i = 0; i < 32; ++i) v[i] = p[(size_t)i * ZR_];
  fft32<SIGN>(v);
  #pragma unroll
  for (int i = 0; i < 32; ++i) p[(size_t)i * ZR_] = v[i];
}

// ---------------- K4/K6: complex FFT along X (in place) ----------------
template<int SIGN>
__global__ __launch_bounds__(256) void fftx_pass(float2* __restrict__ data) {
  int t  = blockIdx.x * blockDim.x + threadIdx.x;    // BV_*D_*Y_*ZR_ = 69632
  int zr = t % ZR_;
  int r  = t / ZR_;
  int y  = r % Y_;
  int bvd = r / Y_;
  float2* p = data + (size_t)bvd * NF_ + (size_t)y * ZR_ + zr;
  float2 v[32];
  #pragma unroll
  for (int i = 0; i < 32; ++i) v[i] = p[(size_t)i * (Y_ * ZR_)];
  fft32<SIGN>(v);
  #pragma unroll
  for (int i = 0; i < 32; ++i) p[(size_t)i * (Y_ * ZR_)] = v[i];
}

// ---------------- K5: frequency-domain Theta/Phi mixing (WMMA + TDM) ----------------
//  Tf[l,h,k] = sum_d Theta[l,h,d] * Xf[d,k]      (real GEMM, re & im planes)
//  Sf[h,k]   = sum_l conj(Phi[l,k]) * Tf[l,h,k]  (VALU fold on accumulators)
#define MXW 4   // waves per block
#define KT  16  // frequency columns per wave
__global__ __launch_bounds__(128) void mix_kernel(const float2* __restrict__ Xf,
                                                  const float*  __restrict__ Theta,
                                                  const float*  __restrict__ PhiRe,
                                                  const float*  __restrict__ PhiIm,
                                                  float2* __restrict__ Sf) {
  __shared__ float  Th[LH_][33];        // 256x32 Theta, padded vs bank conflicts
  __shared__ float2 Xc[MXW][D_][KT];    // TDM-landed tile, interleaved re/im
  __shared__ float  Pre[MXW][L_][KT];
  __shared__ float  Pim[MXW][L_][KT];

  const int tid  = threadIdx.x;
  const int lane = tid & 31;
  const int wave = tid >> 5;
  const int bv   = blockIdx.y;
  const int k0   = (blockIdx.x * MXW + wave) * KT;

  // TDM: DMA the 32(d) x 16(k) float2 tile into LDS (one descriptor per wave)
  const float2* gtile = Xf + (size_t)bv * D_ * NF_ + k0;
  tdm_load_2d((unsigned int)(unsigned long long)&Xc[wave][0][0],
              (unsigned long long)gtile,
              /*data_size=*/3u, /*tile_d0=*/KT, /*tile_d1=*/D_,
              /*tensor_d0=*/(unsigned long long)NF_, /*tensor_d1=*/D_,
              /*stride0=*/(unsigned long long)NF_);

  for (int i = tid; i < LH_ * D_; i += 128) Th[i >> 5][i & 31] = Theta[i];
  for (int i = lane; i < L_ * KT; i += 32) {
    int l = i >> 4, kk = i & 15;
    Pre[wave][l][kk] = PhiRe[(size_t)l * NF_ + k0 + kk];
    Pim[wave][l][kk] = PhiIm[(size_t)l * NF_ + k0 + kk];
  }
  __builtin_amdgcn_s_wait_tensorcnt(0);
  __syncthreads();

  const int ln15 = lane & 15;
  const int lhi  = lane >> 4;
  const v8f zero = {0.f,0.f,0.f,0.f,0.f,0.f,0.f,0.f};
  v8f sre0 = zero, sim0 = zero, sre1 = zero, sim1 = zero;

  #pragma unroll
  for (int mt = 0; mt < 16; ++mt) {             // 16 row-tiles of (l,h)
    v8f tre = zero, tim = zero;
    const int row = mt * 16 + ln15;
    #pragma unroll
    for (int ks = 0; ks < 8; ++ks) {            // K = 32 in steps of 4
      const int koff = ks * 4 + 2 * lhi;
      v2f a;  a.x = Th[row][koff];              a.y = Th[row][koff + 1];
      v2f br; br.x = Xc[wave][koff][ln15].x;    br.y = Xc[wave][koff + 1][ln15].x;
      v2f bi; bi.x = Xc[wave][koff][ln15].y;    bi.y = Xc[wave][koff + 1][ln15].y;
      tre = wmma4(a, br, tre);
      tim = wmma4(a, bi, tim);
    }
    const int l   = mt >> 1;
    const float pre = Pre[wave][l][ln15];
    const float pim = Pim[wave][l][ln15];
    if ((mt & 1) == 0) {
      #pragma unroll
      for (int j = 0; j < 8; ++j) {
        sre0[j] += pre * tre[j] + pim * tim[j];
        sim0[j] += pre * tim[j] - pim * tre[j];
      }
    } else {
      #pragma unroll
      for (int j = 0; j < 8; ++j) {
        sre1[j] += pre * tre[j] + pim * tim[j];
        sim1[j] += pre * tim[j] - pim * tre[j];
      }
    }
  }

  float2* sp = Sf + (size_t)bv * D_ * NF_;
  #pragma unroll
  for (int j = 0; j < 8; ++j) {
    const int h0 = j + 8 * lhi;                 // C layout: vgpr j, lanes 16-31 -> M+8
    sp[(size_t)h0        * NF_ + k0 + ln15] = make_float2(sre0[j], sim0[j]);
    sp[(size_t)(h0 + 16) * NF_ + k0 + ln15] = make_float2(sre1[j], sim1[j]);
  }
}

// ---------------- K8: inverse rFFT along Z (Hermitian) ----------------
__global__ __launch_bounds__(256) void fftz_inv(const float2* __restrict__ Sf,
                                                float* __restrict__ S) {
  int line = blockIdx.x * blockDim.x + threadIdx.x;  // 131072
  const float2* q = Sf + (size_t)line * ZR_;
  float2 v[32];
  #pragma unroll
  for (int k = 0; k < ZR_; ++k) v[k] = q[k];
  #pragma unroll
  for (int k = ZR_; k < 32; ++k) {
    float2 c = v[32 - k];
    v[k] = make_float2(c.x, -c.y);
  }
  fft32<1>(v);
  float* p = S + (size_t)line * Z_;
  const float sc = 1.0f / 32768.0f;                  // 1/N per axis, folded once
  #pragma unroll
  for (int n = 0; n < 32; ++n) p[n] = v[n].x * sc;
}

// ---------------- K9: pointwise MLP (WMMA + TDM) + residual ----------------
__global__ __launch_bounds__(128) void mlp_kernel(const float* __restrict__ S,
                                                  const float* __restrict__ x,
                                                  const float* __restrict__ W1,
                                                  const float* __restrict__ b1,
                                                  const float* __restrict__ W2,
                                                  const float* __restrict__ b2,
                                                  float* __restrict__ out) {
  __shared__ float sW1[64][33];
  __shared__ float sW2[32][65];
  __shared__ float sb1[64];
  __shared__ float sb2[32];
  __shared__ float Stile[MXW][D_][KT];
  __shared__ float Htile[MXW][64][KT];

  const int tid  = threadIdx.x;
  const int lane = tid & 31;
  const int wave = tid >> 5;
  const int bv   = blockIdx.y;
  const int s0   = (blockIdx.x * MXW + wave) * KT;   // spatial tile base

  const float* Sp = S + (size_t)bv * D_ * XYZ_;
  const float* xp = x + (size_t)bv * D_ * XYZ_;
  float*       op = out + (size_t)bv * D_ * XYZ_;

  // TDM: DMA the 32(ch) x 16(spatial) f32 tile into LDS
  tdm_load_2d((unsigned int)(unsigned long long)&Stile[wave][0][0],
              (unsigned long long)(Sp + s0),
              /*data_size=*/2u, /*tile_d0=*/KT, /*tile_d1=*/D_,
              /*tensor_d0=*/(unsigned long long)XYZ_, /*tensor_d1=*/D_,
              /*stride0=*/(unsigned long long)XYZ_);

  for (int i = tid; i < 64 * 32; i += 128) sW1[i >> 5][i & 31] = W1[i];
  for (int i = tid; i < 32 * 64; i += 128) sW2[i >> 6][i & 63] = W2[i];
  if (tid < 64) sb1[tid] = b1[tid];
  if (tid < 32) sb2[tid] = b2[tid];
  __builtin_prefetch(xp + s0, 0, 0);
  __builtin_amdgcn_s_wait_tensorcnt(0);
  __syncthreads();

  const int ln15 = lane & 15;
  const int lhi  = lane >> 4;
  const v8f zero = {0.f,0.f,0.f,0.f,0.f,0.f,0.f,0.f};

  // GEMM1: H(64x16) = W1(64x32) * Stile(32x16); bias + exact-erf GELU
  #pragma unroll
  for (int mt = 0; mt < 4; ++mt) {
    v8f acc = zero;
    #pragma unroll
    for (int ks = 0; ks < 8; ++ks) {
      const int koff = ks * 4 + 2 * lhi;
      v2f a;  a.x = sW1[mt * 16 + ln15][koff];  a.y = sW1[mt * 16 + ln15][koff + 1];
      v2f bb; bb.x = Stile[wave][koff][ln15];   bb.y = Stile[wave][koff + 1][ln15];
      acc = wmma4(a, bb, acc);
    }
    #pragma unroll
    for (int j = 0; j < 8; ++j) {
      const int o = mt * 16 + j + 8 * lhi;
      float hv = acc[j] + sb1[o];
      hv = 0.5f * hv * (1.0f + erff(hv * 0.70710678118654752f));
      Htile[wave][o][ln15] = hv;
    }
  }
  __syncthreads();

  // GEMM2: out(32x16) = W2(32x64) * Htile(64x16) + b2 + residual
  #pragma unroll
  for (int mt = 0; mt < 2; ++mt) {
    v8f acc = zero;
    #pragma unroll
    for (int ks = 0; ks < 16; ++ks) {
      const int koff = ks * 4 + 2 * lhi;
      v2f a;  a.x = sW2[mt * 16 + ln15][koff];  a.y = sW2[mt * 16 + ln15][koff + 1];
      v2f bb; bb.x = Htile[wave][koff][ln15];   bb.y = Htile[wave][koff + 1][ln15];
      acc = wmma4(a, bb, acc);
    }
    #pragma unroll
    for (int j = 0; j < 8; ++j) {
      const int o = mt * 16 + j + 8 * lhi;
      const size_t addr = (size_t)o * XYZ_ + s0 + ln15;
      op[addr] = acc[j] + sb2[o] + xp[addr];
    }
  }
}

// ---------------- host launch ----------------
extern "C" void kernel_launch(void* const* d_in, const int* in_sizes, int n_in,
                              void* d_out, int out_size, void* d_ws, size_t ws_size,
                              hipStream_t stream) {
  const float* x     = (const float*)d_in[0];
  const float* PhiRe = (const float*)d_in[1];
  const float* PhiIm = (const float*)d_in[2];
  const float* Theta = (const float*)d_in[3];
  const float* lnw   = (const float*)d_in[4];
  const float* lnb   = (const float*)d_in[5];
  const float* W1    = (const float*)d_in[6];
  const float* b1    = (const float*)d_in[7];
  const float* W2    = (const float*)d_in[8];
  const float* b2    = (const float*)d_in[9];
  float* out = (float*)d_out;

  char* ws = (char*)d_ws;
  float*  zbuf = (float*)ws;                                    // z, reused as S
  float2* Xf   = (float2*)(ws + NTOT_ * sizeof(float));
  float2* Sf   = (float2*)(ws + NTOT_ * sizeof(float) + NFREQ_ * sizeof(float2));

  // 1) LayerNorm
  ln_kernel<<<dim3(512), dim3(256), 0, stream>>>(x, lnw, lnb, zbuf);
  // 2-4) forward rFFT (Z real->ZR, then complex Y, X)
  fftz_fwd<<<dim3(512), dim3(256), 0, stream>>>(zbuf, Xf);
  ffty_pass<-1><<<dim3(272), dim3(256), 0, stream>>>(Xf);
  fftx_pass<-1><<<dim3(272), dim3(256), 0, stream>>>(Xf);
  // 5) fused Theta/Phi mixing in frequency domain (WMMA, TDM tile feed)
  mix_kernel<<<dim3(NF_ / (MXW * KT), BV_), dim3(128), 0, stream>>>(
      Xf, Theta, PhiRe, PhiIm, Sf);
  // 6-8) inverse FFT (X, Y, then Hermitian Z -> real S)
  fftx_pass<1><<<dim3(272), dim3(256), 0, stream>>>(Sf);
  ffty_pass<1><<<dim3(272), dim3(256), 0, stream>>>(Sf);
  fftz_inv<<<dim3(512), dim3(256), 0, stream>>>(Sf, zbuf);
  // 9) pointwise MLP (WMMA, TDM tile feed) + residual
  mlp_kernel<<<dim3(XYZ_ / (MXW * KT), BV_), dim3(128), 0, stream>>>(
      zbuf, x, W1, b1, W2, b2, out);
}